// ParagraphVectorLoss_32091995636393
// MI455X (gfx1250) — compile-verified
//
#include <hip/hip_runtime.h>
#include <hip/hip_bf16.h>

typedef __attribute__((ext_vector_type(2))) float v2f;
typedef __attribute__((ext_vector_type(8))) float v8f;

#define DIM        128
#define MAXLEN     256
#define NNEG       5
#define BSZ        1024
#define POS_TILES  (MAXLEN / 16)                 // 16
#define NEG_TILES  (MAXLEN * NNEG / 16)          // 80
#define ROW_TILES  (POS_TILES + NEG_TILES)       // 96
#define WPB        8                             // waves per block (wave32)

// numerically stable log_sigmoid(x) = min(x,0) - log1p(exp(-|x|))
__device__ __forceinline__ float log_sigmoid(float x) {
    return fminf(x, 0.0f) - log1pf(__expf(-fabsf(x)));
}

__global__ void pv_init(float* acc) {
    acc[0] = 0.0f;   // sum of log-sigmoids
    acc[1] = 0.0f;   // count of nonzero positive tokens
}

// One wave32 = one tile of 16 tokens of one batch row.
// Dot products done with V_WMMA_F32_16X16X4_F32 (full f32 precision).
__global__ __launch_bounds__(256) void pv_loss_main(
    const float* __restrict__ emb_e,      // [BSZ, DIM]
    const float* __restrict__ emb_table,  // [VOCAB, DIM]
    const int*   __restrict__ token_pos,  // [BSZ, MAXLEN]
    const int*   __restrict__ token_neg,  // [BSZ, MAXLEN*NNEG]
    float* __restrict__ acc)              // acc[0]=sum, acc[1]=cnt
{
    __shared__ float s_sum;
    __shared__ float s_cnt;
    const int tid = threadIdx.x;
    if (tid == 0) { s_sum = 0.0f; s_cnt = 0.0f; }
    __syncthreads();

    const int lane = tid & 31;
    const int wave = tid >> 5;
    const int gwid = blockIdx.x * WPB + wave;        // global wave id
    const int b    = gwid / ROW_TILES;
    const int tile = gwid - b * ROW_TILES;
    const int m    = lane & 15;                      // token row within tile
    const bool hi  = (lane >= 16);                   // lanes 16..31 hold K=2,3
    const bool is_pos = (tile < POS_TILES);

    // gather token id (lanes l and l+16 reference the same token m)
    int tok;
    if (is_pos) tok = token_pos[b * MAXLEN + tile * 16 + m];
    else        tok = token_neg[b * (MAXLEN * NNEG) + (tile - POS_TILES) * 16 + m];

    // A layout (f32 16x4): VGPR0/1 = K {0,1} for lanes 0-15, K {2,3} for lanes 16-31
    const float* arow = emb_table + (size_t)tok * DIM + (hi ? 2 : 0);
    // B (4x16) with every column n replicated = emb_e[b], mirrored lane layout
    const float* brow = emb_e + (size_t)b * DIM + (hi ? 2 : 0);

    v8f c = {};   // 16x16 f32 accumulator; every column holds the same 16 dots
#pragma unroll
    for (int kk = 0; kk < DIM / 4; ++kk) {
        v2f av = *(const v2f*)(arow + 4 * kk);
        v2f bv = *(const v2f*)(brow + 4 * kk);
        // 8 args: (neg_a, A, neg_b, B, c_mod, C, reuse_a, reuse_b)
        c = __builtin_amdgcn_wmma_f32_16x16x4_f32(
                false, av, false, bv, (short)0, c, false, false);
    }

    // Column N=0 lives in lanes 0 (M=0..7 in c[0..7]) and 16 (M=8..15).
    const float sign = is_pos ? 1.0f : -1.0f;
    float local = 0.0f;
    if (m == 0) {
#pragma unroll
        for (int v = 0; v < 8; ++v)
            local += log_sigmoid(sign * c[v]);
    }
    // nonzero-token count (positives only; count each token once via lanes 0-15)
    float cnt = (is_pos && !hi && tok != 0) ? 1.0f : 0.0f;

    // wave32 butterfly reduction
#pragma unroll
    for (int off = 16; off > 0; off >>= 1) {
        local += __shfl_xor(local, off, 32);
        cnt   += __shfl_xor(cnt,   off, 32);
    }
    if (lane == 0) {
        atomicAdd(&s_sum, local);   // ds_add_f32
        atomicAdd(&s_cnt, cnt);
    }
    __syncthreads();
    if (tid == 0) {
        atomicAdd(&acc[0], s_sum);  // one global f32 atomic per block
        atomicAdd(&acc[1], s_cnt);
    }
}

__global__ void pv_finalize(const float* __restrict__ acc, float* __restrict__ out) {
    // n_token = (NNEG + 1) * sum(lens)
    out[0] = -acc[0] / (6.0f * acc[1]);
}

extern "C" void kernel_launch(void* const* d_in, const int* in_sizes, int n_in,
                              void* d_out, int out_size, void* d_ws, size_t ws_size,
                              hipStream_t stream) {
    const float* emb_e     = (const float*)d_in[0];
    const float* emb_table = (const float*)d_in[1];
    const int*   token_pos = (const int*)d_in[2];
    const int*   token_neg = (const int*)d_in[3];
    float* acc = (float*)d_ws;     // 2 floats of scratch
    float* out = (float*)d_out;

    pv_init<<<1, 1, 0, stream>>>(acc);

    const int total_waves = BSZ * ROW_TILES;       // 98304
    const int blocks      = total_waves / WPB;     // 12288
    pv_loss_main<<<blocks, 256, 0, stream>>>(emb_e, emb_table,
                                             token_pos, token_neg, acc);

    pv_finalize<<<1, 1, 0, stream>>>(acc, out);
}